// ComputeAlignmentError_3178275799346
// MI455X (gfx1250) — compile-verified
//
#include <hip/hip_runtime.h>
#include <math.h>

typedef __attribute__((ext_vector_type(2))) float v2f;
typedef __attribute__((ext_vector_type(8))) float v8f;

#define EPS_NORM 1e-8f
#define EPS_ERR  1e-8f

__device__ __forceinline__ void norm3(float& x, float& y, float& z) {
    float nrm = sqrtf(x * x + y * y + z * z);
    float inv = 1.0f / fmaxf(nrm, EPS_NORM);
    x *= inv; y *= inv; z *= inv;
}

// One thread per (batch, frame). Computes pred basis E, negated true basis (-Et),
// and the per-(j,k) column bias Cb = (o_true . e_true_k) - (o_pred . e_pred_k).
__global__ void pae_frame_prep(const float* __restrict__ predF,
                               const float* __restrict__ trueF,
                               float* __restrict__ Ep,    // [bn][9] (k*3+d)
                               float* __restrict__ Etn,   // [bn][9] negated
                               float* __restrict__ Cb,    // [bn][3]
                               int total)
{
    int idx = blockIdx.x * blockDim.x + threadIdx.x;
    if (idx >= total) return;

    float e[2][3][3];
    float s[2][3];
    #pragma unroll
    for (int w = 0; w < 2; ++w) {
        const float* f = (w ? trueF : predF) + (size_t)idx * 9;
        // frames[b,n,d,c]: columns c are points; a=c0, origin=c1, c=c2
        float ox = f[1], oy = f[4], oz = f[7];
        float ax = f[0] - ox, ay = f[3] - oy, az = f[6] - oz;   // a - b
        float cx = f[2] - ox, cy = f[5] - oy, cz = f[8] - oz;   // c - b
        norm3(ax, ay, az);            // w1
        norm3(cx, cy, cz);            // w2
        float e1x = ax + cx, e1y = ay + cy, e1z = az + cz;
        norm3(e1x, e1y, e1z);
        float e2x = cx - ax, e2y = cy - ay, e2z = cz - az;
        norm3(e2x, e2y, e2z);
        float e3x = e1y * e2z - e1z * e2y;
        float e3y = e1z * e2x - e1x * e2z;
        float e3z = e1x * e2y - e1y * e2x;
        e[w][0][0] = e1x; e[w][0][1] = e1y; e[w][0][2] = e1z;
        e[w][1][0] = e2x; e[w][1][1] = e2y; e[w][1][2] = e2z;
        e[w][2][0] = e3x; e[w][2][1] = e3y; e[w][2][2] = e3z;
        #pragma unroll
        for (int k = 0; k < 3; ++k)
            s[w][k] = ox * e[w][k][0] + oy * e[w][k][1] + oz * e[w][k][2];
    }

    float* ep = Ep  + (size_t)idx * 9;
    float* et = Etn + (size_t)idx * 9;
    float* cb = Cb  + (size_t)idx * 3;
    #pragma unroll
    for (int k = 0; k < 3; ++k) {
        #pragma unroll
        for (int d = 0; d < 3; ++d) {
            ep[k * 3 + d] =  e[0][k][d];
            et[k * 3 + d] = -e[1][k][d];   // fold A*(-B) negation here (f32 WMMA has no A/B NEG)
        }
        cb[k] = s[1][k] - s[0][k];         // + (o_true.Et - o_pred.Ep)
    }
}

// One wave per 16x16 output tile. Rows = coord index i, cols = frame index j.
// acc_k = Ap x Bp_k + (At x (-Bt_k) + Cbias_k) via two chained v_wmma_f32_16x16x4_f32.
__global__ void pae_tile_wmma(const float* __restrict__ predC,
                              const float* __restrict__ trueC,
                              const float* __restrict__ Ep,
                              const float* __restrict__ Etn,
                              const float* __restrict__ Cb,
                              float* __restrict__ out,
                              int n, int tpr, int totalTiles)
{
    int tid  = blockIdx.x * blockDim.x + threadIdx.x;
    int wave = tid >> 5;
    int lane = tid & 31;
    if (wave >= totalTiles) return;   // wave-uniform: EXEC stays all-ones past here

    int tilesPerBatch = tpr * tpr;
    int batch = wave / tilesPerBatch;
    int rem   = wave - batch * tilesPerBatch;
    int ti    = rem / tpr;
    int tj    = rem - ti * tpr;
    int i_base = ti << 4;
    int j_base = tj << 4;
    int m    = lane & 15;
    int half = lane >> 4;
    float padSel = half ? 0.0f : 1.0f;    // zero-pad K=3 -> K=4 via cndmask, no branches

    // A-matrix 16x4 f32 layout: lanes0-15 {V0:K0, V1:K1}, lanes16-31 {V0:K2, V1:K3(pad=0)}
    // All loads unconditional; the K=3 pad is applied to the loaded VALUE (v_cndmask),
    // so EXEC stays all-ones everywhere (WMMA requirement, and no exec save/restore churn).
    size_t cbase = ((size_t)batch * n + (size_t)(i_base + m)) * 3;
    size_t dsel  = (size_t)(half << 1);   // 0 or 2
    v2f Ap, At;
    Ap.x = predC[cbase + dsel];
    Ap.y = predC[cbase + 1] * padSel;
    At.x = trueC[cbase + dsel];
    At.y = trueC[cbase + 1] * padSel;

    size_t jidx = (size_t)batch * n + (size_t)(j_base + m);
    const float* ep = Ep  + jidx * 9;
    const float* et = Etn + jidx * 9;
    const float* cb = Cb  + jidx * 3;

    v8f acc[3];
    #pragma unroll
    for (int k = 0; k < 3; ++k) {
        v2f Bp, Bt;   // B 4x16: same K-per-VGPR striping as A, N = lane&15
        Bp.x = ep[k * 3 + dsel];
        Bp.y = ep[k * 3 + 1] * padSel;
        Bt.x = et[k * 3 + dsel];
        Bt.y = et[k * 3 + 1] * padSel;
        float cv = cb[k];            // constant down each column j
        v8f C = {cv, cv, cv, cv, cv, cv, cv, cv};
        v8f t = __builtin_amdgcn_wmma_f32_16x16x4_f32(
            false, At, false, Bt, (short)0, C, false, false);
        acc[k] = __builtin_amdgcn_wmma_f32_16x16x4_f32(
            false, Ap, false, Bp, (short)0, t, false, false);
    }

    // C/D layout: VGPR r -> row r (lanes0-15) / row r+8 (lanes16-31), col = lane&15
    size_t obase = ((size_t)batch * n + (size_t)(i_base + half * 8)) * (size_t)n
                 + (size_t)(j_base + m);
    #pragma unroll
    for (int r = 0; r < 8; ++r) {
        float d0 = acc[0][r], d1 = acc[1][r], d2 = acc[2][r];
        out[obase + (size_t)r * n] = sqrtf(d0 * d0 + d1 * d1 + d2 * d2 + EPS_ERR);
    }
}

extern "C" void kernel_launch(void* const* d_in, const int* in_sizes, int n_in,
                              void* d_out, int out_size, void* d_ws, size_t ws_size,
                              hipStream_t stream) {
    (void)n_in; (void)ws_size;
    const float* predC = (const float*)d_in[0];
    const float* trueC = (const float*)d_in[1];
    const float* predF = (const float*)d_in[2];
    const float* trueF = (const float*)d_in[3];
    float* out = (float*)d_out;

    int bn = in_sizes[0] / 3;                               // b*n = 4096
    int n  = (int)((long long)out_size * 3 / in_sizes[0]);  // 2048
    int b  = bn / n;                                        // 2

    float* Ep  = (float*)d_ws;
    float* Etn = Ep  + (size_t)bn * 9;
    float* Cb  = Etn + (size_t)bn * 9;

    {
        int threads = 256;
        int blocks = (bn + threads - 1) / threads;
        pae_frame_prep<<<blocks, threads, 0, stream>>>(predF, trueF, Ep, Etn, Cb, bn);
    }
    {
        int tpr = n / 16;                            // 128
        int totalTiles = b * tpr * tpr;              // 32768 waves
        int threads = 256;                           // 8 waves / block
        long long totalThreads = (long long)totalTiles * 32;
        int blocks = (int)((totalThreads + threads - 1) / threads);
        pae_tile_wmma<<<blocks, threads, 0, stream>>>(predC, trueC, Ep, Etn, Cb,
                                                      out, n, tpr, totalTiles);
    }
}